// MedianBlur_89876485636252
// MI455X (gfx1250) — compile-verified
//
#include <hip/hip_runtime.h>
#include <stdint.h>

// Iterated 3x3 median blur (edge-padded), 32 x 512 x 512 f32, 10 steps.
// Memory-bound stencil: per step 64MB traffic; 2x32MB ping-pong buffers stay
// resident in the 192MB L2. Data path: per-lane edge-clamped
// GLOBAL_LOAD_ASYNC_TO_LDS_B32 (GVS saddr form, ASYNCcnt-tracked) gathers the
// halo tile into LDS; compute is a wave32 min/max sorting network out of LDS.
// Inactive images (step >= t[b], uniform per block) take a float4 copy path.

#define IMG   512
#define TW    128      // tile width  (pixels per WG)
#define TH    16       // tile height (pixels per WG)
#define HW    (TW + 2) // halo tile width  = 130
#define HH    (TH + 2) // halo tile height = 18
#define HN    (HH * HW)// 2340 halo elements
#define LDSW  132      // padded LDS row stride (floats)
#define NTHR  256      // 8 waves (wave32)
#define FULL  (HN / NTHR)        // 9 full fill rounds
#define TAIL  (HN - FULL * NTHR) // 36 tail lanes

__device__ __forceinline__ void s2(float& a, float& b) {
    float t = fminf(a, b);
    b = fmaxf(a, b);
    a = t;
}

// McGuire 20-exchange median-of-9 network (v_min_num_f32 / v_max_num_f32).
__device__ __forceinline__ float median9(float v0, float v1, float v2,
                                         float v3, float v4, float v5,
                                         float v6, float v7, float v8) {
    // mnmx6(v0..v5)
    s2(v0, v3); s2(v1, v4); s2(v2, v5);
    s2(v0, v1); s2(v0, v2);
    s2(v4, v5); s2(v3, v5);
    // mnmx5(v1,v2,v3,v4,v6)
    s2(v1, v2); s2(v3, v4);
    s2(v1, v3); s2(v1, v6);
    s2(v4, v6); s2(v2, v6);
    // mnmx4(v2,v3,v4,v7)
    s2(v2, v3); s2(v4, v7); s2(v2, v4); s2(v3, v7);
    // mnmx3(v3,v4,v8)
    s2(v4, v8); s2(v3, v8); s2(v3, v4);
    return v4;
}

// One edge-clamped async gather: element e of the halo tile -> LDS.
// GVS form: mem = SGPR64(base) + VGPR32(byte offset); LDS dst = VGPR32 offset.
__device__ __forceinline__ void async_fill_one(
    int e, int x0, int y0, uint64_t img_base, uint32_t lds_base)
{
    const int ly = e / HW;
    const int lx = e - ly * HW;
    const int gx = min(max(x0 - 1 + lx, 0), IMG - 1);
    const int gy = min(max(y0 - 1 + ly, 0), IMG - 1);
    const uint32_t goff  = (uint32_t)(gy * IMG + gx) * 4u;
    const uint32_t laddr = lds_base + (uint32_t)(ly * LDSW + lx) * 4u;
    asm volatile("global_load_async_to_lds_b32 %0, %1, %2"
                 :: "v"(laddr), "v"(goff), "s"(img_base) : "memory");
}

__global__ __launch_bounds__(NTHR) void median_step_kernel(
    const float* __restrict__ src, float* __restrict__ dst,
    const int* __restrict__ t_arr, int step)
{
    __shared__ float tile[HH * LDSW];

    const int tid = threadIdx.x;
    const int b   = blockIdx.z;
    const int x0  = blockIdx.x * TW;
    const int y0  = blockIdx.y * TH;

    const float* img  = src + (size_t)b * (IMG * IMG);
    float*       dimg = dst + (size_t)b * (IMG * IMG);

    // Uniform per-block predicate (scalar load + scalar branch).
    const bool active = step < t_arr[b];

    if (!active) {
        // Passthrough: coalesced float4 copy of the 128x16 tile.
        // 2048 floats = 512 float4 slots; 256 threads x 2 rounds.
        #pragma unroll
        for (int k = 0; k < 2; ++k) {
            const int s   = tid + k * NTHR;       // float4 slot
            const int row = s >> 5;               // 32 slots per 128-wide row
            const int c4  = s & 31;
            const size_t off = (size_t)(y0 + row) * IMG + x0 + c4 * 4;
            *(float4*)(dimg + off) = *(const float4*)(img + off);
        }
        return;
    }

    // LDS byte offset of the tile: flat shared-pointer bits [31:0] are the
    // LDS address (ISA 10.2 aperture mapping) expected by the async VDST.
    const uint32_t lds_base = (uint32_t)(uintptr_t)(void*)&tile[0];
    const uint64_t img_base = (uint64_t)(uintptr_t)img;

    // --- async gather of the edge-clamped halo tile into LDS -------------
    #pragma unroll
    for (int k = 0; k < FULL; ++k)
        async_fill_one(tid + k * NTHR, x0, y0, img_base, lds_base);
    if (tid < TAIL)
        async_fill_one(tid + FULL * NTHR, x0, y0, img_base, lds_base);

#if __has_builtin(__builtin_amdgcn_s_wait_asynccnt)
    __builtin_amdgcn_s_wait_asynccnt(0);
#else
    asm volatile("s_wait_asynccnt 0x0" ::: "memory");
#endif
    __syncthreads();

    // --- compute: each thread sweeps 8 rows of one column ----------------
    const int x  = tid & (TW - 1);   // 0..127
    const int yg = tid >> 7;         // 0..1
    const int r0 = yg * 8;           // first tile row handled by this thread
    const int lx = x + 1;            // halo-local column

    // rolling 3x3 window: halo rows r..r+2 produce output row r
    int base = r0 * LDSW + lx;
    float a0 = tile[base - 1], a1 = tile[base], a2 = tile[base + 1];
    base += LDSW;
    float b0 = tile[base - 1], b1 = tile[base], b2 = tile[base + 1];

    #pragma unroll
    for (int i = 0; i < 8; ++i) {
        base += LDSW;
        const float c0 = tile[base - 1], c1 = tile[base], c2 = tile[base + 1];
        const float m = median9(a0, a1, a2, b0, b1, b2, c0, c1, c2);
        dimg[(size_t)(y0 + r0 + i) * IMG + (x0 + x)] = m;
        a0 = b0; a1 = b1; a2 = b2;
        b0 = c0; b1 = c1; b2 = c2;
    }
}

extern "C" void kernel_launch(void* const* d_in, const int* in_sizes, int n_in,
                              void* d_out, int out_size, void* d_ws, size_t ws_size,
                              hipStream_t stream) {
    (void)n_in; (void)out_size; (void)ws_size;

    const float* x     = (const float*)d_in[0];   // (32,1,512,512) f32
    const int*   t_arr = (const int*)d_in[1];     // (32,) int
    // d_in[2] = kernel_size (3), d_in[3] = num_timesteps (10): fixed by setup.

    float* out = (float*)d_out;
    float* ws  = (float*)d_ws;                    // needs 32 MB scratch

    const int B = in_sizes[1];                    // 32 images
    const int NSTEPS = 10;                        // even -> final step writes d_out

    dim3 grid(IMG / TW, IMG / TH, B);
    dim3 block(NTHR);

    const float* src = x;
    for (int j = 0; j < NSTEPS; ++j) {
        float* dst = (j & 1) ? out : ws;          // j=9 (last) writes d_out
        median_step_kernel<<<grid, block, 0, stream>>>(src, dst, t_arr, j);
        src = dst;
    }
}